// ComposeTransform_22565758173902
// MI455X (gfx1250) — compile-verified
//
#include <hip/hip_runtime.h>

// ComposeTransform on MI455X (gfx1250).
// Shapes fixed by reference: warps [2,128,128,128,3] f32, channel-fastest.
// Bandwidth-bound gather kernel: no WMMA mapping exists (per-lane divergent
// gathers, AI ~0.6 FLOP/B). Optimization = cache-tier control:
//   - NT loads for the streamed warp1, NT stores for the write-once output
//     -> keeps warp2/warp3 gather volumes resident in the 192MB L2.
//   - global_prefetch_b8 one block ahead on the streaming read.
//   - shift/mask index math (dims are powers of two), wave32-friendly blocks.

#define DIMP2 7                    // log2(128)
#define DMAX  127
#define VOX_PER_BATCH (1 << 21)    // 128^3

__device__ __forceinline__ void trilerp3(const float* __restrict__ vol,
                                         float fd, float fh, float fw,
                                         float& r0, float& r1, float& r2) {
  // clip to [0, 127] (matches jnp.clip in reference)
  fd = fminf(fmaxf(fd, 0.0f), 127.0f);
  fh = fminf(fmaxf(fh, 0.0f), 127.0f);
  fw = fminf(fmaxf(fw, 0.0f), 127.0f);

  float d0f = floorf(fd), h0f = floorf(fh), w0f = floorf(fw);
  int d0 = (int)d0f, h0 = (int)h0f, w0 = (int)w0f;
  int d1 = min(d0 + 1, DMAX), h1 = min(h0 + 1, DMAX), w1 = min(w0 + 1, DMAX);

  float td = fd - d0f, th = fh - h0f, tw = fw - w0f;
  float sd = 1.0f - td, sh = 1.0f - th, sw = 1.0f - tw;

  // row bases: voxel index = (d<<14) + (h<<7) + w ; *3 floats per voxel
  int b00 = (d0 << 14) + (h0 << 7);
  int b01 = (d0 << 14) + (h1 << 7);
  int b10 = (d1 << 14) + (h0 << 7);
  int b11 = (d1 << 14) + (h1 << 7);

  int i000 = (b00 + w0) * 3, i001 = (b00 + w1) * 3;
  int i010 = (b01 + w0) * 3, i011 = (b01 + w1) * 3;
  int i100 = (b10 + w0) * 3, i101 = (b10 + w1) * 3;
  int i110 = (b11 + w0) * 3, i111 = (b11 + w1) * 3;

  float wsh = sd * sh, wst = sd * th, wts = td * sh, wtt = td * th;
  float w000 = wsh * sw, w001 = wsh * tw;
  float w010 = wst * sw, w011 = wst * tw;
  float w100 = wts * sw, w101 = wts * tw;
  float w110 = wtt * sw, w111 = wtt * tw;

  r0 = 0.0f; r1 = 0.0f; r2 = 0.0f;
#define ACC(IDX, WGT)                                   \
  { r0 = fmaf((WGT), vol[(IDX) + 0], r0);               \
    r1 = fmaf((WGT), vol[(IDX) + 1], r1);               \
    r2 = fmaf((WGT), vol[(IDX) + 2], r2); }
  ACC(i000, w000) ACC(i001, w001)
  ACC(i010, w010) ACC(i011, w011)
  ACC(i100, w100) ACC(i101, w101)
  ACC(i110, w110) ACC(i111, w111)
#undef ACC
}

__global__ __launch_bounds__(256) void compose_warps_kernel(
    const float* __restrict__ w1, const float* __restrict__ w2,
    const float* __restrict__ w3, float* __restrict__ out, int n_vox_total) {
  int tid = blockIdx.x * blockDim.x + threadIdx.x;   // global voxel id
  if (tid >= n_vox_total) return;

  int b = tid >> 21;                    // batch (128^3 = 2^21 voxels/batch)
  int v = tid & (VOX_PER_BATCH - 1);
  int d = v >> 14;
  int h = (v >> DIMP2) & DMAX;
  int w = v & DMAX;

  long base = (long)tid * 3;
  const float* vol2 = w2 + (long)b * (VOX_PER_BATCH * 3);
  const float* vol3 = w3 + (long)b * (VOX_PER_BATCH * 3);

  // Prefetch the streaming warp1 data one block ahead (global_prefetch_b8).
  __builtin_prefetch(w1 + base + 256 * 3, 0, 3);

  // warp1 is read exactly once -> non-temporal (keep L2 for gather volumes).
  float a0 = __builtin_nontemporal_load(w1 + base + 0);
  float a1 = __builtin_nontemporal_load(w1 + base + 1);
  float a2 = __builtin_nontemporal_load(w1 + base + 2);

  float fd = (float)d, fh = (float)h, fw = (float)w;

  // Stage 1: ab = a + trilerp(b, x + a)
  float s0, s1, s2;
  trilerp3(vol2, fd + a0, fh + a1, fw + a2, s0, s1, s2);
  float u0 = a0 + s0, u1 = a1 + s1, u2 = a2 + s2;

  // Stage 2: out = ab + trilerp(c, x + ab)   (intermediate never hits memory)
  float t0, t1, t2;
  trilerp3(vol3, fd + u0, fh + u1, fw + u2, t0, t1, t2);

  // Write-once output -> non-temporal stores.
  __builtin_nontemporal_store(u0 + t0, out + base + 0);
  __builtin_nontemporal_store(u1 + t1, out + base + 1);
  __builtin_nontemporal_store(u2 + t2, out + base + 2);
}

extern "C" void kernel_launch(void* const* d_in, const int* in_sizes, int n_in,
                              void* d_out, int out_size, void* d_ws, size_t ws_size,
                              hipStream_t stream) {
  const float* w1 = (const float*)d_in[0];
  const float* w2 = (const float*)d_in[1];
  const float* w3 = (const float*)d_in[2];
  float* out = (float*)d_out;

  int n_vox_total = in_sizes[0] / 3;   // 2 * 128^3 = 4,194,304
  const int block = 256;               // 8 wave32 per block
  int grid = (n_vox_total + block - 1) / block;

  compose_warps_kernel<<<grid, block, 0, stream>>>(w1, w2, w3, out, n_vox_total);
}